// MLPDecoder_52802327937710
// MI455X (gfx1250) — compile-verified
//
#include <hip/hip_runtime.h>

typedef __attribute__((ext_vector_type(2))) float v2f;
typedef __attribute__((ext_vector_type(8))) float v8f;

#define IN_DIM 256
#define HIDDEN 64
#define TN 32    // n-rows per block (pairwise kernel)
#define TM 128   // m-cols per block (pairwise kernel)

// ---------------------------------------------------------------------------
// Kernel A: Hout[rows,64] = X[rows,256] @ W[256,64] (+ bias), via
// V_WMMA_F32_16X16X4_F32.  One wave (32 threads) per block; each block
// produces a 16-row x 64-col tile (4 column tiles of 16).
//
// Fragment layouts per CDNA5 ISA 7.12.2 (32-bit, wave32):
//   A 16x4 : lanes 0-15 -> M=lane, VGPR0=K+0, VGPR1=K+1
//            lanes16-31 -> M=lane-16, VGPR0=K+2, VGPR1=K+3
//   B 4x16 : lanes 0-15 -> N=lane, VGPR0=K+0 row, VGPR1=K+1 row
//            lanes16-31 -> N=lane-16, VGPR0=K+2 row, VGPR1=K+3 row
//   C/D    : VGPR r, lanes 0-15 -> (M=r,    N=lane)
//                    lanes16-31 -> (M=r+8,  N=lane-16)
// ---------------------------------------------------------------------------
__global__ __launch_bounds__(32)
void hidden_gemm_wmma(const float* __restrict__ X,    // [rows, IN_DIM]
                      const float* __restrict__ W,    // [IN_DIM, HIDDEN]
                      const float* __restrict__ bias, // [HIDDEN] or nullptr
                      float* __restrict__ Hout)       // [rows, HIDDEN]
{
    const int lane    = threadIdx.x;       // 0..31
    const int half    = lane >> 4;         // 0 | 1
    const int l16     = lane & 15;
    const int koff    = half << 1;         // 0 | 2
    const int rowBase = blockIdx.x * 16;
    const int row     = rowBase + l16;

    v8f acc[4];
    #pragma unroll
    for (int c = 0; c < 4; ++c) {
        const float bv = bias ? bias[c * 16 + l16] : 0.0f;  // col-dependent only
        #pragma unroll
        for (int r = 0; r < 8; ++r) acc[c][r] = bv;
    }

    const float* __restrict__ xrow = X + (size_t)row * IN_DIM;

    for (int k = 0; k < IN_DIM; k += 4) {
        // A fragment: two consecutive K values for this lane's half (8B aligned)
        const v2f a = *(const v2f*)(xrow + k + koff);

        #pragma unroll
        for (int c = 0; c < 4; ++c) {
            const int col = c * 16 + l16;
            v2f b;
            b.x = W[(size_t)(k + koff)     * HIDDEN + col];
            b.y = W[(size_t)(k + koff + 1) * HIDDEN + col];
            acc[c] = __builtin_amdgcn_wmma_f32_16x16x4_f32(
                /*neg_a=*/false, a, /*neg_b=*/false, b,
                /*c_mod=*/(short)0, acc[c],
                /*reuse_a=*/false, /*reuse_b=*/false);
        }
    }

    #pragma unroll
    for (int c = 0; c < 4; ++c) {
        const int col = c * 16 + l16;
        #pragma unroll
        for (int r = 0; r < 8; ++r) {
            const int orow = rowBase + r + half * 8;
            Hout[(size_t)orow * HIDDEN + col] = acc[c][r];
        }
    }
}

// ---------------------------------------------------------------------------
// Kernel B: out[n,m] = sigmoid(b2 + sum_h relu(ag_h[n,h]+ab_h[m,h]) * W2[h])
// Block = 256 threads, tile = TN x TM. Each thread owns a 4(n) x 4(m)
// register micro-tile, with the 4 m-values packed into two float2 lanes so
// the add/fma use packed VOP3P f32 math (v_pk_add_f32 / v_pk_fma_f32):
// per h per thread: 9 LDS dword reads vs 16 pk-ops + 16 v_max -> VALU-bound.
// ---------------------------------------------------------------------------
__global__ __launch_bounds__(256)
void pairwise_relu_dot(const float* __restrict__ agh, // [N, HIDDEN]
                       const float* __restrict__ abh, // [M, HIDDEN]
                       const float* __restrict__ W2,  // [HIDDEN]
                       const float* __restrict__ b2p, // [1]
                       float* __restrict__ out,       // [N, M]
                       int Mtot)
{
    __shared__ float sAg[TN * HIDDEN];   // 8 KB
    __shared__ float sAb[TM * HIDDEN];   // 32 KB
    __shared__ float sW2[HIDDEN];

    const int t     = threadIdx.x;
    const int nBase = blockIdx.y * TN;
    const int mBase = blockIdx.x * TM;

    // Tiles are contiguous row-blocks of [.,HIDDEN] arrays -> coalesced copies.
    for (int i = t; i < TN * HIDDEN; i += 256)
        sAg[i] = agh[(size_t)nBase * HIDDEN + i];
    for (int i = t; i < TM * HIDDEN; i += 256)
        sAb[i] = abh[(size_t)mBase * HIDDEN + i];
    if (t < HIDDEN) sW2[t] = W2[t];
    __syncthreads();

    const int lane = t & 31;
    const int wave = t >> 5;
    const float b2 = b2p[0];

    // acc[i][p]: n-row i, m pair p (p=0 -> {lane, lane+32}, p=1 -> {+64, +96})
    v2f acc[4][2];
    #pragma unroll
    for (int i = 0; i < 4; ++i) {
        acc[i][0] = (v2f){0.0f, 0.0f};
        acc[i][1] = (v2f){0.0f, 0.0f};
    }

    const float* __restrict__ ag0 = &sAg[(wave * 4) * HIDDEN];
    const float* __restrict__ ab0 = &sAb[lane * HIDDEN];
    const v2f zero2 = (v2f){0.0f, 0.0f};

    #pragma unroll 8
    for (int h = 0; h < HIDDEN; ++h) {
        const float w  = sW2[h];
        const v2f  wv  = (v2f){w, w};
        const float a0 = ag0[0 * HIDDEN + h];
        const float a1 = ag0[1 * HIDDEN + h];
        const float a2 = ag0[2 * HIDDEN + h];
        const float a3 = ag0[3 * HIDDEN + h];
        v2f e01, e23;
        e01.x = ab0[0  * HIDDEN + h];
        e01.y = ab0[32 * HIDDEN + h];
        e23.x = ab0[64 * HIDDEN + h];
        e23.y = ab0[96 * HIDDEN + h];
        const float av[4] = {a0, a1, a2, a3};
        #pragma unroll
        for (int i = 0; i < 4; ++i) {
            const v2f ai = (v2f){av[i], av[i]};
            const v2f r0 = __builtin_elementwise_max(ai + e01, zero2); // pk_add + 2x max
            const v2f r1 = __builtin_elementwise_max(ai + e23, zero2);
            acc[i][0] = __builtin_elementwise_fma(r0, wv, acc[i][0]);  // pk_fma
            acc[i][1] = __builtin_elementwise_fma(r1, wv, acc[i][1]);
        }
    }

    #pragma unroll
    for (int i = 0; i < 4; ++i) {
        const int n = nBase + wave * 4 + i;
        #pragma unroll
        for (int p = 0; p < 2; ++p) {
            #pragma unroll
            for (int c = 0; c < 2; ++c) {
                const int m = mBase + lane + (p * 2 + c) * 32;
                const float x = acc[i][p][c] + b2;
                out[(size_t)n * Mtot + m] = 1.0f / (1.0f + __expf(-x));
            }
        }
    }
}

// ---------------------------------------------------------------------------
extern "C" void kernel_launch(void* const* d_in, const int* in_sizes, int n_in,
                              void* d_out, int out_size, void* d_ws, size_t ws_size,
                              hipStream_t stream) {
    const float* ag = (const float*)d_in[0];  // [N, 256]
    const float* ab = (const float*)d_in[1];  // [M, 256]
    const float* W1 = (const float*)d_in[2];  // [512, 64] row-major
    const float* b1 = (const float*)d_in[3];  // [64]
    const float* W2 = (const float*)d_in[4];  // [64]
    const float* b2 = (const float*)d_in[5];  // scalar (1 elem)
    float* out = (float*)d_out;

    const int N = in_sizes[0] / IN_DIM;       // 2048
    const int M = in_sizes[1] / IN_DIM;       // 2048

    float* agh = (float*)d_ws;                // [N, HIDDEN]
    float* abh = agh + (size_t)N * HIDDEN;    // [M, HIDDEN]

    // Phase 1: hidden activations via fp32 WMMA.
    hidden_gemm_wmma<<<N / 16, 32, 0, stream>>>(ag, W1, b1, agh);
    hidden_gemm_wmma<<<M / 16, 32, 0, stream>>>(ab, W1 + (size_t)IN_DIM * HIDDEN,
                                                nullptr, abh);

    // Phase 2: pairwise relu-dot + sigmoid (VALU-bound, packed f32 math).
    dim3 grid(M / TM, N / TN);
    pairwise_relu_dot<<<grid, 256, 0, stream>>>(agh, abh, W2, b2, out, M);
}